// Actor_Net_PNAConv_Model_54271206752463
// MI455X (gfx1250) — compile-verified
//
#include <hip/hip_runtime.h>
#include <hip/hip_bf16.h>

#define NNODES 40000
#define NEDGES 640000
#define NPER   1000
#define BGR    40
#define DM     64
#define SNORM  0.031622776601683794f
#define LEAK   0.01f

typedef __attribute__((ext_vector_type(2))) float v2f;
typedef __attribute__((ext_vector_type(8))) float v8f;
typedef __attribute__((ext_vector_type(4))) unsigned int u32x4;
typedef __attribute__((ext_vector_type(8))) int i32x8;
typedef __attribute__((ext_vector_type(4))) int i32x4;

__device__ inline v8f wmma4(v2f a, v2f b, v8f c) {
    // V_WMMA_F32_16X16X4_F32 : D = A(16x4,f32) x B(4x16,f32) + C(16x16,f32)
    return __builtin_amdgcn_wmma_f32_16x16x4_f32(
        false, a, false, b, (short)0, c, false, false);
}

// Tensor Data Mover: DMA a 2D tile (f32 elements) global -> LDS.
// tile_d1 rows of tile_d0 elements; row y at gsrc + y*stride0 elements;
// packed into LDS at lds_byte_off with row pitch tile_d0 elements.
__device__ inline void tdm_load_2d(const float* gsrc, unsigned lds_byte_off,
                                   int tile_d0, int tile_d1,
                                   int tensor_d0, int tensor_d1, int stride0)
{
    unsigned long long ga = (unsigned long long)(void*)gsrc;
    u32x4 g0;
    g0[0] = 1u;                                     // count=1 (valid user D#)
    g0[1] = lds_byte_off;                           // lds_addr
    g0[2] = (unsigned)(ga & 0xffffffffull);         // global_addr[31:0]
    g0[3] = (unsigned)((ga >> 32) & 0x1ffffffull) | (2u << 30); // [56:32] | type=2
    i32x8 g1;
    g1[0] = 2 << 16;                                // data_size = 4 bytes
    g1[1] = (int)(((unsigned)tensor_d0 & 0xffffu) << 16);          // dim0[15:0]
    g1[2] = (int)((((unsigned)tensor_d0 >> 16) & 0xffffu) |
                  (((unsigned)tensor_d1 & 0xffffu) << 16));        // dim0 hi | dim1 lo
    g1[3] = (int)((((unsigned)tensor_d1 >> 16) & 0xffffu) |
                  (((unsigned)tile_d0 & 0xffffu) << 16));          // dim1 hi | tile_dim0
    g1[4] = (int)((unsigned)tile_d1 & 0xffffu);                    // tile_dim1 (dim2=0)
    g1[5] = stride0;                                // tensor_dim0_stride[31:0]
    g1[6] = 0;
    g1[7] = 0;
    i32x4 z4 = {0, 0, 0, 0};
    i32x8 z8 = {0, 0, 0, 0, 0, 0, 0, 0};
    __builtin_amdgcn_tensor_load_to_lds(g0, g1, z4, z4, z8, 0);
}

__device__ inline void atomicMaxFloat(float* addr, float val) {
    if (val >= 0.0f) atomicMax((int*)addr, __float_as_int(val));
    else             atomicMin((unsigned int*)addr, __float_as_uint(val));
}

// ---------------------------------------------------------------- utilities
__global__ void k_fill(float* __restrict__ p, float v, int n) {
    int i = blockIdx.x * blockDim.x + threadIdx.x;
    if (i < n) p[i] = v;
}

__global__ void k_deg(const int* __restrict__ dst, float* __restrict__ deg, int E) {
    int e = blockIdx.x * blockDim.x + threadIdx.x;
    if (e < E) atomicAdd(&deg[dst[e]], 1.0f);
}

__global__ void k_build_h0(const float* __restrict__ nf, float* __restrict__ h, int N) {
    int idx = blockIdx.x * blockDim.x + threadIdx.x;
    if (idx >= N * DM) return;
    int i = idx / DM, k = idx % DM;
    h[idx] = (k < 15) ? nf[i * 16 + k] : 0.0f;
}

__global__ void k_build_t(const float* __restrict__ h2, const float* __restrict__ nf,
                          float* __restrict__ t, int N) {
    int idx = blockIdx.x * blockDim.x + threadIdx.x;
    if (idx >= N * DM) return;
    int i = idx / DM, k = idx % DM;
    float v = 0.0f;
    if (k < 8)       v = h2[i * DM + k];
    else if (k < 12) v = nf[i * 16 + 15];
    t[idx] = v;
}

// ------------------------------------------------- edge message + aggregate
// block = 128 threads (4 waves), 32 edges per block.
// msg = leaky_relu([h[src],h[dst],e] @ Mw^T + Mb); scatter sum/sumsq/max by dst.
// Mw (din x F, row-major) is DMA'd into LDS by the TDM while threads gather A.
// Inner loop is guard-free: A is zero-padded in K, and pad columns (colg>=din)
// produce garbage that the epilogue never stores.
__global__ __launch_bounds__(128) void k_msg_agg(
    const float* __restrict__ h, const float* __restrict__ ef,
    const int* __restrict__ src, const int* __restrict__ dst,
    const float* __restrict__ Mw, const float* __restrict__ Mb,
    float* __restrict__ sum, float* __restrict__ sumsq, float* __restrict__ mx,
    int E, int din, int F, int Fpad, int dinpad)
{
    extern __shared__ float lds[];
    float* As = lds;                            // 32 x Fpad
    float* Bw = lds + 32 * Fpad;                // dinpad x F slack (TDM fills din x F)
    int*   es = (int*)(Bw + dinpad * F + 4);    // 32 src indices
    int*   ed = es + 32;                        // 32 dst indices

    int e0 = blockIdx.x * 32;
    int tid = threadIdx.x;

    if (tid < 32) {
        // async DMA of the weight tile, overlapped with the A gather below
        tdm_load_2d(Mw, __builtin_amdgcn_groupstaticsize() + 32u * Fpad * 4u,
                    F, din, F, din, F);
        int e = e0 + tid;
        es[tid] = (e < E) ? src[e] : 0;
        ed[tid] = (e < E) ? dst[e] : 0;
    }
    __syncthreads();

    for (int idx = tid; idx < 32 * Fpad; idx += 128) {
        int r = idx / Fpad, k = idx % Fpad;
        float v = 0.0f;
        if (e0 + r < E) {
            if (k < din)            v = h[es[r] * DM + k];
            else if (k < 2 * din)   v = h[ed[r] * DM + (k - din)];
            else if (k < F)         v = ef[(e0 + r) * 2 + (k - 2 * din)];
        }
        As[idx] = v;
    }
    if (tid < 32) __builtin_amdgcn_s_wait_tensorcnt(0);
    __syncthreads();

    int wave = tid >> 5, lane = tid & 31;
    int ncols = dinpad / 16;
    int ntiles = 2 * ncols;
    int kk = (lane >> 4) * 2;
    for (int t = wave; t < ntiles; t += 4) {
        int tr = t / ncols, tc = t % ncols;
        int row0 = tr * 16, col0 = tc * 16;
        int ar = row0 + (lane & 15);
        int colg = col0 + (lane & 15);
        const float* arow = As + ar * Fpad;
        const float* brow = Bw + colg * F;
        v8f acc = {};
        for (int k0 = 0; k0 < Fpad; k0 += 4) {
            v2f a, b;
            a.x = arow[k0 + kk];
            a.y = arow[k0 + kk + 1];
            b.x = brow[k0 + kk];
            b.y = brow[k0 + kk + 1];
            acc = wmma4(a, b, acc);
        }
        if (colg < din) {
            float bias = Mb[colg];
            int rbase = row0 + 8 * (lane >> 4);
#pragma unroll
            for (int j = 0; j < 8; ++j) {
                int r = rbase + j;
                if (e0 + r < E) {
                    float v = acc[j] + bias;
                    v = (v > 0.0f) ? v : LEAK * v;
                    int d = ed[r];
                    atomicAdd(&sum[d * DM + colg], v);
                    atomicAdd(&sumsq[d * DM + colg], v * v);
                    atomicMaxFloat(&mx[d * DM + colg], v);
                }
            }
        }
    }
}

// ------------------------------------------------------ node update (U GEMM)
// u = ([h, mean, max, sum, std] @ Uw^T + Ub) * snorm ; accumulate BN stats.
// Uw is K-chunked through LDS via TDM (chunk = KC x doutpad <= 20 KB).
__global__ __launch_bounds__(128) void k_update(
    const float* __restrict__ h, const float* __restrict__ sum,
    const float* __restrict__ sumsq, const float* __restrict__ mxv,
    const float* __restrict__ deg, const float* __restrict__ Uw,
    const float* __restrict__ Ub, float* __restrict__ u,
    float* __restrict__ bnstat, int N, int din, int dout,
    int K5, int K5pad, int doutpad, int KC)
{
    extern __shared__ float lds[];
    float* As = lds;                 // 32 x K5pad
    float* Bs = lds + 32 * K5pad;    // KC x doutpad (row n at n*kw)
    unsigned bs_off = __builtin_amdgcn_groupstaticsize() + 32u * K5pad * 4u;

    int i0 = blockIdx.x * 32;
    int tid = threadIdx.x;

    for (int idx = tid; idx < 32 * K5pad; idx += 128) {
        int r = idx / K5pad, k = idx % K5pad;
        int i = i0 + r;
        float v = 0.0f;
        if (i < N && k < K5) {
            int sect = k / din;
            int j = k - sect * din;
            if (sect == 0) {
                v = h[i * DM + j];
            } else {
                float dg = deg[i];
                float degc = fmaxf(dg, 1.0f);
                float s = sum[i * DM + j];
                if (sect == 1)      v = s / degc;
                else if (sect == 2) v = (dg > 0.0f) ? mxv[i * DM + j] : 0.0f;
                else if (sect == 3) v = s;
                else {
                    float mean = s / degc;
                    float ms = sumsq[i * DM + j] / degc;
                    float var = fmaxf(ms - mean * mean, 0.0f);
                    v = (dg > 0.0f) ? sqrtf(var + 1e-30f) : 0.0f;
                }
            }
        }
        As[idx] = v;
    }
    __syncthreads();

    int wave = tid >> 5, lane = tid & 31;
    int ncols = doutpad / 16;
    int ntiles = 2 * ncols;
    int kk = (lane >> 4) * 2;
    v8f acc[2] = {};
    for (int kc = 0; kc < K5pad; kc += KC) {
        int kw = (K5pad - kc < KC) ? (K5pad - kc) : KC;
        if (tid < 32) {
            tdm_load_2d(Uw + kc, bs_off, kw, dout, K5 - kc, dout, K5);
            __builtin_amdgcn_s_wait_tensorcnt(0);
        }
        __syncthreads();
#pragma unroll
        for (int ti = 0; ti < 2; ++ti) {
            int t = wave + 4 * ti;
            if (t < ntiles) {
                int tr = t / ncols, tc = t % ncols;
                int ar = tr * 16 + (lane & 15);
                int colg = tc * 16 + (lane & 15);
                const float* arow = As + ar * K5pad + kc;
                const float* brow = Bs + colg * kw;
                v8f a8 = acc[ti];
                for (int k0 = 0; k0 < kw; k0 += 4) {
                    v2f a, b;
                    a.x = arow[k0 + kk];
                    a.y = arow[k0 + kk + 1];
                    b.x = brow[k0 + kk];
                    b.y = brow[k0 + kk + 1];
                    a8 = wmma4(a, b, a8);
                }
                acc[ti] = a8;
            }
        }
        __syncthreads();
    }

#pragma unroll
    for (int ti = 0; ti < 2; ++ti) {
        int t = wave + 4 * ti;
        if (t < ntiles) {
            int tr = t / ncols, tc = t % ncols;
            int colg = tc * 16 + (lane & 15);
            if (colg < dout) {
                float bias = Ub[colg];
                float bs = 0.0f, bq = 0.0f;
                int rbase = i0 + tr * 16 + 8 * (lane >> 4);
#pragma unroll
                for (int j = 0; j < 8; ++j) {
                    int i = rbase + j;
                    if (i < N) {
                        float val = (acc[ti][j] + bias) * SNORM;
                        u[i * DM + colg] = val;
                        bs += val;
                        bq += val * val;
                    }
                }
                atomicAdd(&bnstat[colg], bs);
                atomicAdd(&bnstat[DM + colg], bq);
            }
        }
    }
}

// ------------------------------------------ batchnorm + X GEMM + activation
__global__ __launch_bounds__(128) void k_bnx(
    const float* __restrict__ u, const float* __restrict__ hprev,
    const float* __restrict__ bnstat, const float* __restrict__ bng,
    const float* __restrict__ bnb, const float* __restrict__ Xw,
    const float* __restrict__ Xb, float* __restrict__ hnext,
    int N, int dout, int Kxpad, int doutpad, int residual, int reluflag)
{
    extern __shared__ float lds[];
    float* As = lds;                 // 32 x Kxpad
    float* Bw = lds + 32 * Kxpad;    // doutpad x dout slack (TDM fills dout x dout)
    int i0 = blockIdx.x * 32;
    int tid = threadIdx.x;
    float Nf = (float)N;

    if (tid < 32) {
        tdm_load_2d(Xw, __builtin_amdgcn_groupstaticsize() + 32u * Kxpad * 4u,
                    dout, dout, dout, dout, dout);
    }

    for (int idx = tid; idx < 32 * Kxpad; idx += 128) {
        int r = idx / Kxpad, k = idx % Kxpad;
        int i = i0 + r;
        float v = 0.0f;
        if (i < N && k < dout) {
            float mu = bnstat[k] / Nf;
            float var = bnstat[DM + k] / Nf - mu * mu;
            float inv = rsqrtf(var + 1e-5f);
            v = (u[i * DM + k] - mu) * inv * bng[k] + bnb[k];
        }
        As[idx] = v;
    }
    if (tid < 32) __builtin_amdgcn_s_wait_tensorcnt(0);
    __syncthreads();

    int wave = tid >> 5, lane = tid & 31;
    int ncols = doutpad / 16;
    int ntiles = 2 * ncols;
    int kk = (lane >> 4) * 2;
    for (int t = wave; t < ntiles; t += 4) {
        int tr = t / ncols, tc = t % ncols;
        int row0 = tr * 16, col0 = tc * 16;
        int ar = row0 + (lane & 15);
        int colg = col0 + (lane & 15);
        const float* arow = As + ar * Kxpad;
        const float* brow = Bw + colg * dout;
        v8f acc = {};
        for (int k0 = 0; k0 < Kxpad; k0 += 4) {
            v2f a, b;
            a.x = arow[k0 + kk];
            a.y = arow[k0 + kk + 1];
            b.x = brow[k0 + kk];
            b.y = brow[k0 + kk + 1];
            acc = wmma4(a, b, acc);
        }
        int rbase = i0 + row0 + 8 * (lane >> 4);
#pragma unroll
        for (int j = 0; j < 8; ++j) {
            int i = rbase + j;
            if (i < N) {
                float val = 0.0f;
                if (colg < dout) {
                    val = acc[j] + Xb[colg];
                    val = (val > 0.0f) ? val : LEAK * val;
                    if (residual) val += hprev[i * DM + colg];
                    if (reluflag) val = fmaxf(val, 0.0f);
                }
                hnext[i * DM + colg] = val;
            }
        }
    }
}

// ----------------------------------------------------------- graph softmax
__global__ __launch_bounds__(256) void k_softmax(const float* __restrict__ hfin,
                                                 float* __restrict__ out)
{
    __shared__ float red[256];
    int b = blockIdx.x, tid = threadIdx.x;
    const float* base = hfin + (size_t)b * NPER * DM;

    float m = -__builtin_inff();
    for (int j = tid; j < NPER; j += 256) m = fmaxf(m, base[j * DM]);
    red[tid] = m; __syncthreads();
    for (int s = 128; s > 0; s >>= 1) {
        if (tid < s) red[tid] = fmaxf(red[tid], red[tid + s]);
        __syncthreads();
    }
    m = red[0]; __syncthreads();

    float sm = 0.0f;
    for (int j = tid; j < NPER; j += 256) sm += expf(base[j * DM] - m);
    red[tid] = sm; __syncthreads();
    for (int s = 128; s > 0; s >>= 1) {
        if (tid < s) red[tid] += red[tid + s];
        __syncthreads();
    }
    float rinv = 1.0f / red[0];

    for (int j = tid; j < NPER; j += 256)
        out[b * NPER + j] = expf(base[j * DM] - m) * rinv;
}

// ------------------------------------------------------------------- driver
struct LCfg { int din, dout, residual, relu; };
static const LCfg LC[6] = {
    {15, 64, 0, 1}, {64, 64, 1, 1}, {64, 8, 0, 1},
    {12, 64, 0, 1}, {64, 64, 1, 1}, {64, 1, 0, 0}
};

extern "C" void kernel_launch(void* const* d_in, const int* in_sizes, int n_in,
                              void* d_out, int out_size, void* d_ws, size_t ws_size,
                              hipStream_t stream)
{
    const float* nf  = (const float*)d_in[0];
    const float* ef  = (const float*)d_in[1];
    const int*   src = (const int*)d_in[2];
    const int*   dst = (const int*)d_in[3];
    // params flattened: layer-major, dict keys sorted:
    // 0:Mb 1:Mw 2:Ub 3:Uw 4:Xb 5:Xw 6:bn_b 7:bn_g
    auto P = [&](int layer, int which) -> const float* {
        return (const float*)d_in[4 + layer * 8 + which];
    };

    const int N = NNODES, E = NEDGES;
    float* ws     = (float*)d_ws;
    float* hA     = ws;
    float* hB     = hA + (size_t)N * DM;
    float* uB     = hB + (size_t)N * DM;
    float* sum    = uB + (size_t)N * DM;
    float* sumsq  = sum + (size_t)N * DM;
    float* mx     = sumsq + (size_t)N * DM;
    float* deg    = mx + (size_t)N * DM;
    float* bnstat = deg + N;

    const int tpb = 256;
    k_fill<<<(N + tpb - 1) / tpb, tpb, 0, stream>>>(deg, 0.0f, N);
    k_deg<<<(E + tpb - 1) / tpb, tpb, 0, stream>>>(dst, deg, E);
    k_build_h0<<<(N * DM + tpb - 1) / tpb, tpb, 0, stream>>>(nf, hA, N);

    float* hin = hA;
    float* hout = hB;
    for (int l = 0; l < 6; ++l) {
        if (l == 3) {
            // t = concat([h2 (8), tile(ns_s, 4)]) -> 12 dims
            k_build_t<<<(N * DM + tpb - 1) / tpb, tpb, 0, stream>>>(hin, nf, hout, N);
            float* tmp = hin; hin = hout; hout = tmp;
        }
        const int din = LC[l].din, dout = LC[l].dout;
        const int F = 2 * din + 2, Fpad = (F + 3) & ~3;
        const int dinpad = (din + 15) & ~15;
        const int K5 = 5 * din, K5pad = (K5 + 3) & ~3;
        const int Kxpad = (dout + 3) & ~3;
        const int doutpad = (dout + 15) & ~15;
        const int KC = (K5pad < 80) ? K5pad : 80;

        k_fill<<<(N * DM + tpb - 1) / tpb, tpb, 0, stream>>>(sum, 0.0f, N * DM);
        k_fill<<<(N * DM + tpb - 1) / tpb, tpb, 0, stream>>>(sumsq, 0.0f, N * DM);
        k_fill<<<(N * DM + tpb - 1) / tpb, tpb, 0, stream>>>(mx, -__builtin_inff(), N * DM);
        k_fill<<<1, 256, 0, stream>>>(bnstat, 0.0f, 2 * DM);

        size_t shm_msg = (size_t)(32 * Fpad + dinpad * F + 4) * 4 + 64 * 4;
        k_msg_agg<<<(E + 31) / 32, 128, shm_msg, stream>>>(
            hin, ef, src, dst, P(l, 1), P(l, 0),
            sum, sumsq, mx, E, din, F, Fpad, dinpad);

        size_t shm_upd = (size_t)(32 * K5pad + KC * doutpad) * 4;
        k_update<<<(N + 31) / 32, 128, shm_upd, stream>>>(
            hin, sum, sumsq, mx, deg, P(l, 3), P(l, 2),
            uB, bnstat, N, din, dout, K5, K5pad, doutpad, KC);

        size_t shm_bnx = (size_t)(32 * Kxpad + doutpad * dout + 4) * 4;
        k_bnx<<<(N + 31) / 32, 128, shm_bnx, stream>>>(
            uB, hin, bnstat, P(l, 7), P(l, 6), P(l, 5), P(l, 4),
            hout, N, dout, Kxpad, doutpad, LC[l].residual, LC[l].relu);

        float* tmp = hin; hin = hout; hout = tmp;
    }

    k_softmax<<<BGR, 256, 0, stream>>>(hin, (float*)d_out);
}